// TimeEmbedding_26276609917323
// MI455X (gfx1250) — compile-verified
//
#include <hip/hip_runtime.h>

// out[i, :] = W[elt[i], :] + b   (N x 128, fp32)
// Strategy: store-bandwidth-bound (HBM floor ~11.5us for 268MB of NT stores).
// W (5.2MB) stays resident in the 192MB L2; gathers are L2 hits.
// Bias add is folded into v_wmma_f32_16x16x4_f32: D = ones_col x b_row + C.

typedef __attribute__((ext_vector_type(2))) float v2f;
typedef __attribute__((ext_vector_type(4))) float v4f;
typedef __attribute__((ext_vector_type(8))) float v8f;

#define EMB 128

__global__ __launch_bounds__(256) void TimeEmbedding_gather_wmma(
    const int* __restrict__ elt, const float* __restrict__ W,
    const float* __restrict__ b, float* __restrict__ out, int ngroups) {
  const int lane = threadIdx.x & 31;
  const int wib = threadIdx.x >> 5;                 // wave in block
  const int wavesPerBlock = blockDim.x >> 5;        // 8
  const int gwave = blockIdx.x * wavesPerBlock + wib;
  const int nwaves = gridDim.x * wavesPerBlock;
  const int hi = lane >> 4;                         // half-wave select
  const int lo = lane & 15;

  // A operand (16x4 f32): column K=0 is all 1.0, everything else 0.
  // Layout: VGPR0 lanes0-15 = K=0 (M=lane); all other slots zeroed.
  v2f A;
  A.x = (lane < 16) ? 1.0f : 0.0f;
  A.y = 0.0f;

  // B operands (4x16 f32), one per 16-col tile: row K=0 = bias chunk, rest 0.
  v2f Bop[8];
#pragma unroll
  for (int t = 0; t < 8; ++t) {
    float bv = b[t * 16 + lo];
    Bop[t].x = (lane < 16) ? bv : 0.0f;
    Bop[t].y = 0.0f;
  }

  // Wave-uniform grid-stride loop over groups of 16 tokens -> EXEC stays all-1s.
  for (int g = gwave; g < ngroups; g += nwaves) {
    const int t0 = __builtin_amdgcn_readfirstlane(g) * 16;
    const int* ip = elt + t0;                       // uniform ptr -> s_load
    int idx[16];
#pragma unroll
    for (int i = 0; i < 16; ++i) idx[i] = ip[i];

    // Prefetch next group's indices (global_prefetch_b8).
    __builtin_prefetch(elt + t0 + 16 * nwaves, 0, 1);

    // Gather W rows straight into WMMA C layout:
    // C vgpr r, lane: element (M = r + 8*hi, N = 16*t + lo).
    v8f c[8];
#pragma unroll
    for (int r = 0; r < 8; ++r) {
      const int row = hi ? idx[r + 8] : idx[r];
      const float* arow = W + (size_t)row * EMB + lo;
#pragma unroll
      for (int t = 0; t < 8; ++t) c[t][r] = arow[t * 16];
    }

    // D = A x B + C  ==  C + broadcast(bias chunk), exact in fp32.
#pragma unroll
    for (int t = 0; t < 8; ++t)
      c[t] = __builtin_amdgcn_wmma_f32_16x16x4_f32(
          /*neg_a=*/false, A, /*neg_b=*/false, Bop[t],
          /*c_mod=*/(short)0, c[t], /*reuse_a=*/false, /*reuse_b=*/false);

    // Non-temporal stores: stream 268MB out without evicting W from L2.
#pragma unroll
    for (int r = 0; r < 8; ++r) {
      float* orow = out + (size_t)(t0 + r + 8 * hi) * EMB + lo;
#pragma unroll
      for (int t = 0; t < 8; ++t)
        __builtin_nontemporal_store(c[t][r], orow + t * 16);
    }
  }
}

// Generic tail (N % 16 != 0): one wave per token, float4 per lane.
__global__ void TimeEmbedding_tail(const int* __restrict__ elt,
                                   const float* __restrict__ W,
                                   const float* __restrict__ b,
                                   float* __restrict__ out, int start, int n) {
  int i = blockIdx.x * blockDim.x + threadIdx.x;
  int tok = start + (i >> 5);
  if (tok >= n) return;
  int c4 = (i & 31) * 4;
  int row = elt[tok];
  v4f w = *(const v4f*)(W + (size_t)row * EMB + c4);
  v4f bb = *(const v4f*)(b + c4);
  v4f o = w + bb;
  __builtin_nontemporal_store(o, (v4f*)(out + (size_t)tok * EMB + c4));
}

extern "C" void kernel_launch(void* const* d_in, const int* in_sizes, int n_in,
                              void* d_out, int out_size, void* d_ws, size_t ws_size,
                              hipStream_t stream) {
  const int* elt = (const int*)d_in[0];     // [N] token ids
  const float* W = (const float*)d_in[1];   // [10080, 128]
  const float* b = (const float*)d_in[2];   // [128]
  float* out = (float*)d_out;               // [N, 128]

  const int n = in_sizes[0];
  const int ngroups = n / 16;
  const int tail = n - ngroups * 16;

  if (ngroups > 0) {
    // 512 blocks x 256 threads = 4096 waves; 32768 groups -> 8 groups/wave.
    TimeEmbedding_gather_wmma<<<512, 256, 0, stream>>>(elt, W, b, out, ngroups);
  }
  if (tail > 0) {
    int threads = tail * 32;
    TimeEmbedding_tail<<<(threads + 255) / 256, 256, 0, stream>>>(
        elt, W, b, out, ngroups * 16, n);
  }
}